// RelationshipConsistencyLoss_35175782154851
// MI455X (gfx1250) — compile-verified
//
#include <hip/hip_runtime.h>
#include <hip/hip_bf16.h>
#include <math.h>

typedef __attribute__((ext_vector_type(2)))  float v2f;
typedef __attribute__((ext_vector_type(8)))  float v8f;

#define NBLOCKS  1024
#define NTHREADS 256

// Numerically-stable BCE-with-logits term, matches the jnp formula exactly:
//   max(s,0) - s*y + log1p(exp(-|s|))
__device__ __forceinline__ float bce_term(float s, float y) {
    return fmaxf(s, 0.0f) - s * y + log1pf(expf(-fabsf(s)));
}

// ---------------------------------------------------------------------------
// Stage 1: streaming per-edge loss, one partial sum per block (deterministic).
// Memory-bound: 12B/edge streamed (b128-vectorized) + 2 L2-resident gathers
// + LDS adjacency lookup per edge.
// ---------------------------------------------------------------------------
__global__ __launch_bounds__(NTHREADS) void edge_loss_partial(
    const int*   __restrict__ node_cls,   // [n_nodes] class id per node
    const float* __restrict__ scores,     // [n_edges] logits
    const int*   __restrict__ eidx,       // [2, n_edges] flat (row0=src, row1=dst)
    const float* __restrict__ adj,        // [ncls, ncls] 0/1 validity
    float*       __restrict__ partials,   // [gridDim.x] block partial sums
    int n_edges, int ncls)
{
    __shared__ float s_adj[256];          // up to 16x16 classes
    const int nadj = ncls * ncls;
    for (int i = threadIdx.x; i < nadj; i += blockDim.x) s_adj[i] = adj[i];
    __syncthreads();

    const int tid      = blockIdx.x * blockDim.x + threadIdx.x;
    const int nthreads = gridDim.x * blockDim.x;

    float acc = 0.0f;

    if ((n_edges & 3) == 0) {
        // Vectorized path: 4 edges / iteration, 3x global_load_b128 per iter.
        const int     n4      = n_edges >> 2;
        const float4* scores4 = (const float4*)scores;
        const int4*   src4    = (const int4*)eidx;
        const int4*   dst4    = (const int4*)(eidx + n_edges);
        for (int i = tid; i < n4; i += nthreads) {
            const float4 s = scores4[i];
            const int4   u = src4[i];
            const int4   v = dst4[i];
            acc += bce_term(s.x, s_adj[node_cls[u.x] * ncls + node_cls[v.x]]);
            acc += bce_term(s.y, s_adj[node_cls[u.y] * ncls + node_cls[v.y]]);
            acc += bce_term(s.z, s_adj[node_cls[u.z] * ncls + node_cls[v.z]]);
            acc += bce_term(s.w, s_adj[node_cls[u.w] * ncls + node_cls[v.w]]);
        }
    } else {
        // Generic scalar path.
        for (int e = tid; e < n_edges; e += nthreads) {
            const float y = s_adj[node_cls[eidx[e]] * ncls
                                  + node_cls[eidx[n_edges + e]]];
            acc += bce_term(scores[e], y);
        }
    }

    __shared__ float red[NTHREADS];
    red[threadIdx.x] = acc;
    __syncthreads();
    for (int off = NTHREADS >> 1; off > 0; off >>= 1) {
        if (threadIdx.x < off) red[threadIdx.x] += red[threadIdx.x + off];
        __syncthreads();
    }
    if (threadIdx.x == 0) partials[blockIdx.x] = red[0];
}

// ---------------------------------------------------------------------------
// Stage 2: single-wave (wave32) WMMA reduction of block partials.
// V_WMMA_F32_16X16X4_F32 with A == all-ones:
//   D[m][n] = sum_k B[k][n] + C[m][n]
// so ANY bijective packing of the 64 floats into B sums correctly; f32
// accumulation through C across tiles. Row 0 of D (lanes 0..15, c[0]) holds
// the 16 column sums. Caller guarantees n is a multiple of 64, so loads are
// unconditional b64 — no EXEC masking around the WMMA.
// ---------------------------------------------------------------------------
__global__ __launch_bounds__(32) void wmma_sum_reduce(
    const float* __restrict__ partials, float* __restrict__ out,
    int n, float scale)
{
    const int lane = threadIdx.x;          // 0..31, EXEC all-ones
    v8f c = {};
    v2f a; a.x = 1.0f; a.y = 1.0f;         // A(16x4) == all ones

    const float2* p2    = (const float2*)partials;
    const int     tiles = n >> 6;          // 64 floats per WMMA tile
    for (int t = 0; t < tiles; ++t) {
        const float2 d = p2[t * 32 + lane];
        v2f b; b.x = d.x; b.y = d.y;
        c = __builtin_amdgcn_wmma_f32_16x16x4_f32(
                /*neg_a=*/false, a, /*neg_b=*/false, b,
                /*c_mod=*/(short)0, c, /*reuse_a=*/false, /*reuse_b=*/false);
    }

    // c[0] on lanes 0..15 = column sums (lanes 16..31 hold the identical
    // M=8 row). Reduce the 16-lane group via xor masks 8,4,2,1 (wave32).
    float v = c[0];
    v += __shfl_xor(v, 8, 32);
    v += __shfl_xor(v, 4, 32);
    v += __shfl_xor(v, 2, 32);
    v += __shfl_xor(v, 1, 32);
    if (lane == 0) out[0] = v * scale;
}

// ---------------------------------------------------------------------------
extern "C" void kernel_launch(void* const* d_in, const int* in_sizes, int n_in,
                              void* d_out, int out_size, void* d_ws, size_t ws_size,
                              hipStream_t stream)
{
    const int*   node_cls = (const int*)  d_in[0];   // node_classes (int)
    const float* scores   = (const float*)d_in[1];   // edge_scores  (f32)
    const int*   eidx     = (const int*)  d_in[2];   // edge_indices (int, [2,E])
    const float* adj      = (const float*)d_in[3];   // valid_adjacency (f32, CxC)
    float*       out      = (float*)d_out;
    float*       partials = (float*)d_ws;

    const int n_edges = in_sizes[1];                 // E (scores are [E,1] flat)
    const int nadj    = in_sizes[3];
    int ncls = 1;
    while (ncls * ncls < nadj) ++ncls;               // C = 12 for the reference

    // Block count: multiple of 64 so the WMMA reduce needs no bounds checks.
    int nblocks = NBLOCKS;
    if ((size_t)nblocks * sizeof(float) > ws_size) {
        nblocks = (int)(ws_size / sizeof(float)) & ~63;
        if (nblocks < 64) nblocks = 64;              // ws is always >= 256B
    }

    edge_loss_partial<<<nblocks, NTHREADS, 0, stream>>>(
        node_cls, scores, eidx, adj, partials, n_edges, ncls);

    wmma_sum_reduce<<<1, 32, 0, stream>>>(
        partials, out, nblocks, 1.0f / (float)n_edges);
}